// GeneralLoss_11819749998868
// MI455X (gfx1250) — compile-verified
//
#include <hip/hip_runtime.h>
#include <math.h>

#define BB 32
#define PP 32768
#define GG 32
#define THRESH_F 0.35f

typedef float v2f __attribute__((ext_vector_type(2)));
typedef float v8f __attribute__((ext_vector_type(8)));
typedef unsigned int v4u __attribute__((ext_vector_type(4)));
typedef int v8i __attribute__((ext_vector_type(8)));
typedef int v4i __attribute__((ext_vector_type(4)));

// ---------------------------------------------------------------------------
// Wave-level sum on the WMMA pipe: D = A(16x4 f32) * ones(4x16) + 0.
// With a=(v,0): rowsum(m) = v(lane m) + v(lane m+16); summing c[0..7] gives
// half-wave partials, one shfl_xor(16) completes the 32-lane sum.
// ---------------------------------------------------------------------------
__device__ __forceinline__ float wave_sum_wmma(float v) {
  v2f a; a.x = v;     a.y = 0.0f;
  v2f o; o.x = 1.0f;  o.y = 1.0f;
  v8f c;
#pragma unroll
  for (int i = 0; i < 8; ++i) c[i] = 0.0f;
  c = __builtin_amdgcn_wmma_f32_16x16x4_f32(false, a, false, o, (short)0, c,
                                            false, false);
  float s = c[0] + c[1] + c[2] + c[3] + c[4] + c[5] + c[6] + c[7];
  return s + __shfl_xor(s, 16, 32);
}

// Block-wide sum (blockDim.x == 1024 => 32 waves). Result valid in wave 0.
__device__ __forceinline__ float block_sum_wmma(float v, float* sbuf) {
  float ws = wave_sum_wmma(v);
  int wave = threadIdx.x >> 5;
  int lane = threadIdx.x & 31;
  __syncthreads();                 // protect sbuf reuse
  if (lane == 0) sbuf[wave] = ws;
  __syncthreads();
  float tot = 0.0f;
  if (wave == 0) tot = wave_sum_wmma(sbuf[lane]);
  return tot;
}

// ---------------------------------------------------------------------------
__global__ void init_kernel(unsigned long long* packed) {
  int i = blockIdx.x * blockDim.x + threadIdx.x;
  if (i < BB * GG) packed[i] = 0ull;
}

// Phase A: IoU match. grid = (PP/256, BB), block = 256.
__global__ void match_kernel(const float* __restrict__ priorbox,
                             const float* __restrict__ targets,
                             float* __restrict__ bt_ov,
                             int* __restrict__ bt_idx,
                             unsigned long long* __restrict__ packed) {
  __shared__ float ga[GG][4];     // gts in swapped [y1,x1,y2,x2] order
  __shared__ float garea[GG];
  __shared__ unsigned long long sbest[GG];
  const int b = blockIdx.y;
  const int t = threadIdx.x;
  const int p = blockIdx.x * blockDim.x + t;

  if (t < GG) {
    const float* tg = targets + (b * GG + t) * 5;
    float x1 = tg[0], y1 = tg[1], x2 = tg[2], y2 = tg[3];
    ga[t][0] = y1; ga[t][1] = x1; ga[t][2] = y2; ga[t][3] = x2;
    garea[t] = (y2 - y1) * (x2 - x1);   // (a2-a0)*(a3-a1)
    sbest[t] = 0ull;
  }
  __syncthreads();

  const float* pr = priorbox + p * 4;
  float cx = pr[0], cy = pr[1], w = pr[2], h = pr[3];
  float bx1 = cx - 0.5f * w, by1 = cy - 0.5f * h;
  float bx2 = cx + 0.5f * w, by2 = cy + 0.5f * h;
  float area_b = (bx2 - bx1) * (by2 - by1);

  float best_ov = -1.0f;
  int best_g = 0;
#pragma unroll 4
  for (int g = 0; g < GG; ++g) {
    float ltx = fmaxf(ga[g][0], bx1);
    float lty = fmaxf(ga[g][1], by1);
    float rbx = fminf(ga[g][2], bx2);
    float rby = fminf(ga[g][3], by2);
    float iw = fmaxf(rbx - ltx, 0.0f);
    float ih = fmaxf(rby - lty, 0.0f);
    float inter = iw * ih;
    float ov = inter / (garea[g] + area_b - inter);
    if (ov > best_ov) { best_ov = ov; best_g = g; }  // first-max tie-break
    // pack (ov_bits, ~p): larger ov wins, ties -> smaller p (first occurrence)
    unsigned long long key =
        ((unsigned long long)__float_as_uint(ov) << 32) |
        (unsigned int)(~(unsigned int)p);
    atomicMax(&sbest[g], key);
  }
  bt_ov[b * PP + p] = best_ov;
  bt_idx[b * PP + p] = best_g;
  __syncthreads();
  if (t < GG) atomicMax(&packed[b * GG + t], sbest[t]);
}

// Phase B: override best prior per gt (sequential in g -> last g wins,
// matching JAX .at[].set scatter order). 1 block, 32 threads (one per b).
__global__ void override_kernel(float* __restrict__ bt_ov,
                                int* __restrict__ bt_idx,
                                const unsigned long long* __restrict__ packed) {
  int b = threadIdx.x;
  if (b < BB) {
    for (int g = 0; g < GG; ++g) {
      unsigned long long key = packed[b * GG + g];
      unsigned int p = ~(unsigned int)(key & 0xffffffffull);
      bt_ov[b * PP + p] = 2.0f;
      bt_idx[b * PP + p] = g;
    }
  }
}

// Phase C: encode + smooth-L1 + CE. grid = BB, block = 1024.
__global__ void lossc_kernel(const float* __restrict__ loc_preds,
                             const float* __restrict__ cls_preds,
                             const float* __restrict__ priorbox,
                             const float* __restrict__ targets,
                             const float* __restrict__ bt_ov,
                             const int* __restrict__ bt_idx,
                             float* __restrict__ f_loss,
                             float* __restrict__ f_locsum,
                             float* __restrict__ f_ceposs,
                             int* __restrict__ i_npos) {
  __shared__ float sbuf[32];
  const int b = blockIdx.x;
  float loc_local = 0.0f, cep_local = 0.0f, np_local = 0.0f;

  for (int p = threadIdx.x; p < PP; p += 1024) {
    const int gi = bt_idx[b * PP + p];
    const float ov = bt_ov[b * PP + p];
    const float* tg = targets + (b * GG + gi) * 5;
    float label = tg[4];
    float conf = (ov < THRESH_F) ? 0.0f : label;
    bool pos = conf > 0.0f;
    bool ign = conf < 0.0f;
    int ci = ign ? 0 : (int)conf;

    // matched gts in swapped order [y1,x1,y2,x2]
    float a0 = tg[1], a1 = tg[0], a2 = tg[3], a3 = tg[2];
    const float* pr = priorbox + p * 4;
    const float* lp = loc_preds + ((size_t)b * PP + p) * 4;
    const float* cp = cls_preds + ((size_t)b * PP + p) * 2;
    __builtin_prefetch(lp + 4096, 0, 0);   // global_prefetch_b8

    float mcx = 0.5f * (a0 + a2), mcy = 0.5f * (a1 + a3);
    float mw = a2 - a0, mh = a3 - a1;
    float g0 = (mcx - pr[0]) / (0.1f * pr[2]);
    float g1 = (mcy - pr[1]) / (0.1f * pr[3]);
    float g2 = logf(mw / pr[2]) * 5.0f;    // /VAR1 (0.2)
    float g3 = logf(mh / pr[3]) * 5.0f;

    float sl = 0.0f;
    float gv[4] = {g0, g1, g2, g3};
#pragma unroll
    for (int i = 0; i < 4; ++i) {
      float d = lp[i] - gv[i];
      float ad = fabsf(d);
      sl += (ad < 1.0f) ? 0.5f * d * d : (ad - 0.5f);
    }
    if (pos) loc_local += sl;

    float c0 = cp[0], c1 = cp[1];
    float m = fmaxf(c0, c1);
    float lse = m + logf(expf(c0 - m) + expf(c1 - m));
    float picked = ci ? c1 : c0;
    float ce = lse - picked;
    if (pos) { cep_local += ce; np_local += 1.0f; }
    f_loss[b * PP + p] = (pos || ign) ? 0.0f : ce;
  }

  float locT = block_sum_wmma(loc_local, sbuf);
  float cepT = block_sum_wmma(cep_local, sbuf);
  float npT  = block_sum_wmma(np_local, sbuf);
  if (threadIdx.x == 0) {
    f_locsum[b] = locT;
    f_ceposs[b] = cepT;
    i_npos[b] = (int)(npT + 0.5f);
  }
}

// Phase D: hard-negative mining. The per-batch 128 KB loss array is DMA'd
// into LDS once by the Tensor Data Mover (TENSOR_LOAD_TO_LDS, TENSORcnt),
// then all 32 scan passes (31-step bitwise binary search for the k-th
// largest + final sum) run out of LDS. grid = BB, block = 1024.
__global__ void negmine_kernel(const float* __restrict__ f_loss,
                               const float* __restrict__ f_ceposs,
                               const int* __restrict__ i_npos,
                               float* __restrict__ f_clssum) {
  __shared__ float sloss[PP];      // 128 KB tile (<= 320 KB WGP LDS)
  __shared__ float sbuf[32];
  __shared__ float sh_tot;
  const int b = blockIdx.x;
  const float* base = f_loss + (size_t)b * PP;

  if (threadIdx.x < 32) {          // wave 0 issues the TDM DMA (EXEC ignored)
    unsigned long long ga = (unsigned long long)(size_t)base;
    unsigned int lds_off = (unsigned int)(size_t)(void*)&sloss[0];
    // D# group 0: count=1, lds_addr, global_addr[56:0], type=2
    v4u g0;
    g0[0] = 1u;
    g0[1] = lds_off;
    g0[2] = (unsigned int)(ga & 0xffffffffull);
    g0[3] = (unsigned int)((ga >> 32) & 0x01ffffffull) | 0x80000000u;
    // D# group 1: data_size=4B; tensor_dim0=32768; tensor_dim1=1;
    // tile_dim0=32768; tile_dim1=1; tensor_dim0_stride=32768
    v8i g1;
    g1[0] = 0x00020000;        // wg_mask=0, data_size=2 (4B)
    g1[1] = (int)0x80000000;   // tensor_dim0 (bits[79:48]) low part = 0x8000
    g1[2] = 0x00010000;        // tensor_dim0 hi=0, tensor_dim1 lo=1
    g1[3] = (int)0x80000000;   // tensor_dim1 hi=0, tile_dim0=0x8000
    g1[4] = 0x00000001;        // tile_dim1=1, tile_dim2=0
    g1[5] = 0x00008000;        // tensor_dim0_stride lo = 32768
    g1[6] = 0;                 // stride hi / dim1_stride lo
    g1[7] = 0;
    v4i gz4; gz4[0] = 0; gz4[1] = 0; gz4[2] = 0; gz4[3] = 0;  // groups 2/3
    v8i gz8;
#pragma unroll
    for (int i = 0; i < 8; ++i) gz8[i] = 0;
    // clang-23 6-arg form: (v4u g0, v8i g1, v4i g2, v4i g3, v8i, i32 cpol)
    __builtin_amdgcn_tensor_load_to_lds(g0, g1, gz4, gz4, gz8, 0);
    __builtin_amdgcn_s_wait_tensorcnt(0);
  }
  __syncthreads();                 // publish LDS tile to all 32 waves

  const int np = i_npos[b];
  int k = 3 * np;
  if (k > PP - 1) k = PP - 1;

  float cls_extra = 0.0f;
  if (k > 0) {
    const float kf = (float)k;
    unsigned int lo = 0u;
    for (int bit = 30; bit >= 0; --bit) {
      unsigned int cand = lo | (1u << bit);
      float cnt = 0.0f;
      for (int p = threadIdx.x; p < PP; p += 1024)
        cnt += (__float_as_uint(sloss[p]) >= cand) ? 1.0f : 0.0f;
      float tot = block_sum_wmma(cnt, sbuf);   // exact: integer-valued f32
      if (threadIdx.x == 0) sh_tot = tot;
      __syncthreads();
      if (sh_tot >= kf) lo = cand;             // uniform decision
    }
    const float t = __uint_as_float(lo);       // k-th largest value
    float sgt = 0.0f, cgt = 0.0f;
    for (int p = threadIdx.x; p < PP; p += 1024) {
      float v = sloss[p];
      if (v > t) { sgt += v; cgt += 1.0f; }
    }
    float SG = block_sum_wmma(sgt, sbuf);
    float CG = block_sum_wmma(cgt, sbuf);
    if (threadIdx.x == 0) cls_extra = SG + (kf - CG) * t;  // ties at t
  }
  if (threadIdx.x == 0) f_clssum[b] = f_ceposs[b] + cls_extra;
}

// Final combine. 1 block, 32 threads (one wave -> EXEC all ones for WMMA).
__global__ void final_kernel(const float* __restrict__ f_locsum,
                             const float* __restrict__ f_clssum,
                             const int* __restrict__ i_npos,
                             float* __restrict__ out) {
  int b = threadIdx.x;
  float locT = wave_sum_wmma(f_locsum[b]);
  float clsT = wave_sum_wmma(f_clssum[b]);
  float npT  = wave_sum_wmma((float)i_npos[b]);
  if (b == 0) {
    float N = (npT > 0.0f) ? npT : (float)BB;
    out[0] = (locT + clsT) / N;
  }
}

// ---------------------------------------------------------------------------
extern "C" void kernel_launch(void* const* d_in, const int* in_sizes, int n_in,
                              void* d_out, int out_size, void* d_ws,
                              size_t ws_size, hipStream_t stream) {
  const float* loc_preds = (const float*)d_in[0];  // B*P*4
  const float* cls_preds = (const float*)d_in[1];  // B*P*2
  const float* priorbox  = (const float*)d_in[2];  // P*4
  const float* targets   = (const float*)d_in[3];  // B*G*5
  float* out = (float*)d_out;

  char* ws = (char*)d_ws;
  const size_t BP = (size_t)BB * PP;
  float* bt_ov  = (float*)ws;                        // BP floats
  int*   bt_idx = (int*)(ws + 4 * BP);               // BP ints
  float* f_loss = (float*)(ws + 8 * BP);             // BP floats
  unsigned long long* packed =
      (unsigned long long*)(ws + 12 * BP);           // B*G u64
  float* f_locsum = (float*)(ws + 12 * BP + 8ull * BB * GG);
  float* f_ceposs = f_locsum + BB;
  float* f_clssum = f_ceposs + BB;
  int*   i_npos   = (int*)(f_clssum + BB);

  init_kernel<<<1, BB * GG, 0, stream>>>(packed);
  dim3 gA(PP / 256, BB);
  match_kernel<<<gA, 256, 0, stream>>>(priorbox, targets, bt_ov, bt_idx,
                                       packed);
  override_kernel<<<1, BB, 0, stream>>>(bt_ov, bt_idx, packed);
  lossc_kernel<<<BB, 1024, 0, stream>>>(loc_preds, cls_preds, priorbox,
                                        targets, bt_ov, bt_idx, f_loss,
                                        f_locsum, f_ceposs, i_npos);
  negmine_kernel<<<BB, 1024, 0, stream>>>(f_loss, f_ceposs, i_npos, f_clssum);
  final_kernel<<<1, 32, 0, stream>>>(f_locsum, f_clssum, i_npos, out);
}